// EmbeddingModel_61503931678942
// MI455X (gfx1250) — compile-verified
//
#include <hip/hip_runtime.h>
#include <stdint.h>

// Gather + top-k: logits = W[input_ids]; top_k(logits, 256) sorted descending.
// One workgroup per token row (B*S = 16384 rows).
//   Stage 1: 8KB W row -> LDS via CDNA5 async global->LDS B128 copies (ASYNCcnt).
//   Stage 2: pack 64-bit keys: (monotonic(f32) << 32) | ~index.
//            larger key == larger value, ties -> smaller index (jax top_k order).
//   Stage 3: bitonic-sort 8 chunks of 256 descending (36 passes @ 2048 wide).
//   Stage 4: 3 prune-merge rounds: A[i]=max(A[i],B[255-i]) (bitonic top-256 of
//            A∪B), then 8-pass bitonic merge. Active width 1024 -> 512 -> 256.
//   Stage 5: emit kv[0..k) : value re-read from staged row, index = ~low32.
// ~90K LDS touches/row vs ~135K for a full 2048 sort.

#define HIDDEN   2048
#define CHUNK    256
#define NTHREADS 256

__global__ __launch_bounds__(NTHREADS) void gather_topk_kernel(
    const int* __restrict__ ids,     // [n_rows] token ids
    const float* __restrict__ W,     // [VOCAB, HIDDEN]
    float* __restrict__ out_vals,    // [n_rows, k]
    int*   __restrict__ out_idx,     // [n_rows, k]
    int k)
{
    __shared__ __align__(16) float vals[HIDDEN];           // staged W row (8KB)
    __shared__ unsigned long long  kv[HIDDEN];             // sort keys (16KB)

    const unsigned tid = threadIdx.x;
    const unsigned row = blockIdx.x;

    const int      r       = ids[row];
    const uint64_t rowbase = (uint64_t)(uintptr_t)(W + (size_t)r * HIDDEN);
    // Low 32 bits of the flat shared-aperture address are the LDS byte offset.
    const unsigned ldsbase = (unsigned)(uintptr_t)&vals[0];

    // ---- Stage 1: async copy 8KB row -> LDS (2x B128 per lane, GVS mode) ----
    {
        unsigned g0 = tid * 16u;          // byte offset into the row
        unsigned l0 = ldsbase + g0;       // LDS byte address (in VGPR)
        asm volatile("global_load_async_to_lds_b128 %0, %1, %2"
                     :: "v"(l0), "v"(g0), "s"(rowbase) : "memory");
        unsigned g1 = g0 + 4096u;
        unsigned l1 = ldsbase + g1;
        asm volatile("global_load_async_to_lds_b128 %0, %1, %2"
                     :: "v"(l1), "v"(g1), "s"(rowbase) : "memory");
    }
    asm volatile("s_wait_asynccnt 0" ::: "memory");   // our wave's copies landed
    __syncthreads();                                  // all waves' copies landed

    // ---- Stage 2: build keys ------------------------------------------------
    for (unsigned i = tid; i < HIDDEN; i += NTHREADS) {
        unsigned u = __float_as_uint(vals[i]);
        unsigned m = (u & 0x80000000u) ? ~u : (u ^ 0x80000000u); // order-preserving
        kv[i] = ((unsigned long long)m << 32) | (unsigned long long)(~i);
    }
    __syncthreads();

    // ---- Stage 3: bitonic-sort each 256-chunk DESCENDING --------------------
    // asc(loc) = ((loc & size) != 0); loc = i & 255. At size==256 the &255
    // kills the bit -> whole chunk descending on the final merge stage.
    for (unsigned size = 2; size <= CHUNK; size <<= 1) {
        for (unsigned stride = size >> 1; stride > 0; stride >>= 1) {
            for (unsigned i = tid; i < HIDDEN; i += NTHREADS) {
                unsigned l = i ^ stride;          // partner stays inside chunk
                if (l > i) {
                    unsigned long long a = kv[i];
                    unsigned long long b = kv[l];
                    bool asc = ((i & size & (CHUNK - 1u)) != 0u);
                    if (asc ? (a > b) : (a < b)) { kv[i] = b; kv[l] = a; }
                }
            }
            __syncthreads();
        }
    }

    // ---- Stage 4: 3 rounds of prune-merge (keep top-256 per pair) -----------
    for (unsigned rnd = 0; rnd < 3; ++rnd) {
        const unsigned half   = CHUNK << rnd;      // A base -> B base offset
        const unsigned span   = (CHUNK * 2u) << rnd; // distance between A bases
        const unsigned active = HIDDEN >> (1u + rnd); // 1024, 512, 256

        // select: A[i] = max(A[i], B[255-i])  (B is read-only, A owned per-i)
        for (unsigned t = tid; t < active; t += NTHREADS) {
            unsigned p = t >> 8, i = t & (CHUNK - 1u);
            unsigned a0 = p * span;
            unsigned long long a = kv[a0 + i];
            unsigned long long b = kv[a0 + half + (CHUNK - 1u) - i];
            kv[a0 + i] = (a > b) ? a : b;          // now bitonic top-256
        }
        __syncthreads();

        // bitonic merge each surviving 256-list descending
        for (unsigned stride = CHUNK >> 1; stride > 0; stride >>= 1) {
            for (unsigned t = tid; t < active; t += NTHREADS) {
                unsigned p = t >> 8, i = t & (CHUNK - 1u);
                unsigned l = i ^ stride;
                if (l > i) {
                    unsigned base = p * span;
                    unsigned long long a = kv[base + i];
                    unsigned long long b = kv[base + l];
                    if (a < b) { kv[base + i] = b; kv[base + l] = a; }
                }
            }
            __syncthreads();
        }
    }

    // ---- Stage 5: emit top-k (kv[0..k) is sorted descending) ----------------
    const unsigned kk = ((unsigned)k < (unsigned)CHUNK) ? (unsigned)k : (unsigned)CHUNK;
    for (unsigned i = tid; i < kk; i += NTHREADS) {
        unsigned long long e = kv[i];
        unsigned idx = ~(unsigned)e;               // low 32 bits were ~index
        out_vals[(size_t)row * (unsigned)k + i] = vals[idx];
        out_idx [(size_t)row * (unsigned)k + i] = (int)idx;
    }
}

extern "C" void kernel_launch(void* const* d_in, const int* in_sizes, int n_in,
                              void* d_out, int out_size, void* d_ws, size_t ws_size,
                              hipStream_t stream) {
    (void)n_in; (void)d_ws; (void)ws_size;
    const int*   ids = (const int*)d_in[0];    // input_ids, flattened [B*S]
    const float* W   = (const float*)d_in[1];  // [VOCAB, HIDDEN]
    const int n_rows = in_sizes[0];            // B*S
    // d_out = values [n_rows,k] (f32) then indices [n_rows,k] (i32), concatenated.
    const int k = out_size / (2 * n_rows);     // = TOPK (256)
    float* out_vals = (float*)d_out;
    int*   out_idx  = (int*)d_out + (size_t)n_rows * (size_t)k;

    gather_topk_kernel<<<n_rows, NTHREADS, 0, stream>>>(ids, W, out_vals, out_idx, k);
}